// FMSynth_32573031973087
// MI455X (gfx1250) — compile-verified
//
#include <hip/hip_runtime.h>
#include <stdint.h>

#define TPB      256
#define NBLK     64
#define SR_INV   (1.0f / 16000.0f)
#define TWO_PI   6.28318530717958647692f
#define INV_2PI  0.15915494309189533577f

typedef unsigned int v4u __attribute__((ext_vector_type(4)));
typedef int          v8i __attribute__((ext_vector_type(8)));
typedef int          v4i __attribute__((ext_vector_type(4)));

__global__ __launch_bounds__(TPB)
void fm_synth_kernel(const float* __restrict__ fm_params,
                     const float* __restrict__ f0_hz,
                     const float* __restrict__ phase_state,
                     float* __restrict__ audio,
                     float* __restrict__ phase_end,
                     int n)
{
    __shared__ float lds_fm[TPB * 6];

    const int tid = threadIdx.x;
    const int b   = blockIdx.x * TPB + tid;

    float p[6];

#if __has_builtin(__builtin_amdgcn_tensor_load_to_lds) && __has_builtin(__builtin_amdgcn_s_wait_tensorcnt)
    {
        // Each wave DMAs its 32 voices x 6 floats (768 B) of fm_params into its
        // LDS slice via the Tensor Data Mover, then reads back only its own slice
        // (no workgroup barrier needed).
        const int      wid      = __builtin_amdgcn_readfirstlane((int)(tid >> 5));
        const uint64_t gaddr    = (uint64_t)(uintptr_t)fm_params +
                                  ((uint64_t)blockIdx.x * TPB + (uint64_t)wid * 32u) * 24u;
        const uint32_t lds_byte = (uint32_t)(uintptr_t)&lds_fm[wid * 32 * 6];
        const uint32_t nelem    = 32u * 6u;   // 192 f32 per wave

        // D# group 0: count=1 valid, lds_addr, global_addr[56:0], type=2
        v4u g0;
        g0.x = 1u;
        g0.y = lds_byte;
        g0.z = (uint32_t)gaddr;
        g0.w = ((uint32_t)(gaddr >> 32) & 0x01FFFFFFu) | 0x80000000u;

        // D# group 1: data_size=4B, tensor_dim0=192, tensor_dim1=1,
        // tile_dim0=192, tensor_dim0_stride=192 (1-D contiguous tile)
        v8i g1;
        g1[0] = (int)(2u << 16);
        g1[1] = (int)(nelem << 16);
        g1[2] = (int)(1u << 16);
        g1[3] = (int)(nelem << 16);
        g1[4] = 0;
        g1[5] = (int)nelem;
        g1[6] = 0;
        g1[7] = 0;

        v4i gz = {0, 0, 0, 0};
#if defined(__clang_major__) && (__clang_major__ >= 23)
        v8i gz8 = {0, 0, 0, 0, 0, 0, 0, 0};
        __builtin_amdgcn_tensor_load_to_lds(g0, g1, gz, gz, gz8, 0);
#else
        __builtin_amdgcn_tensor_load_to_lds(g0, g1, gz, gz, 0);
#endif
        __builtin_amdgcn_s_wait_tensorcnt(0);

        // Launder so the compiler cannot assume LDS is unwritten.
        const float* pf = &lds_fm[tid * 6];
        asm volatile("" : "+v"(pf) :: "memory");
        #pragma unroll
        for (int k = 0; k < 6; ++k) p[k] = pf[k];
    }
#else
    {
        const float* gp = fm_params + (size_t)b * 6u;
        #pragma unroll
        for (int k = 0; k < 6; ++k) p[k] = gp[k];
    }
#endif

    if (b >= n) return;

    const float  f0 = fmaxf(f0_hz[b], 1.0f);
    const float4 ps = ((const float4*)phase_state)[b];

    // Track phase in *revolutions* so v_sin_f32 is used directly (no 2*pi mul,
    // no software range reduction). Feedback/mod coefficients pre-scaled by 1/2pi.
    const float r1   = fmaf(p[1], 15.75f, 0.25f);
    const float fb1  = p[2] * (0.95f * INV_2PI);
    const float d2   = p[3] * (10.0f * INV_2PI);
    const float r2   = fmaf(p[4], 15.75f, 0.25f);
    const float fb2  = p[5] * (0.95f * INV_2PI);

    const float inc1 = f0 * r1 * SR_INV;   // revolutions per sample
    const float inc2 = f0 * r2 * SR_INV;

    float ph1 = ps.x * INV_2PI;
    float ph2 = ps.y * INV_2PI;
    float l1  = ps.z;
    float l2  = ps.w;

    float smp[NBLK];
    float sumsq = 0.0f;

    #pragma unroll
    for (int t = 0; t < NBLK; ++t) {
        const float o1 = __builtin_amdgcn_sinf(fmaf(fb1, l1, ph1));
        const float o2 = __builtin_amdgcn_sinf(fmaf(fb2, l2, fmaf(d2, o1, ph2)));
        smp[t] = o2;
        sumsq  = fmaf(o2, o2, sumsq);
        l1 = o1;
        l2 = o2;
        ph1 += inc1;
        ph2 += inc2;
    }

    const float scale = __builtin_amdgcn_rsqf(fmaf(sumsq, 1.0f / (float)NBLK, 1e-5f));

    float4* __restrict__ orow = (float4*)(audio + (size_t)b * NBLK);
    #pragma unroll
    for (int q = 0; q < NBLK / 4; ++q) {
        float4 v;
        v.x = smp[4 * q + 0] * scale;
        v.y = smp[4 * q + 1] * scale;
        v.z = smp[4 * q + 2] * scale;
        v.w = smp[4 * q + 3] * scale;
        orow[q] = v;
    }

    // phase_end: (start + 64*inc_rad) mod 2pi, plus final feedback state
    const float x1 = fmaf((float)NBLK * TWO_PI, inc1, ps.x);
    const float x2 = fmaf((float)NBLK * TWO_PI, inc2, ps.y);
    const float m1 = fmaf(-floorf(x1 * INV_2PI), TWO_PI, x1);
    const float m2 = fmaf(-floorf(x2 * INV_2PI), TWO_PI, x2);
    float4 pe;
    pe.x = m1; pe.y = m2; pe.z = l1; pe.w = l2;
    ((float4*)phase_end)[b] = pe;
}

extern "C" void kernel_launch(void* const* d_in, const int* in_sizes, int n_in,
                              void* d_out, int out_size, void* d_ws, size_t ws_size,
                              hipStream_t stream) {
    const float* fm = (const float*)d_in[0];   // (B, 1, 6) f32
    const float* f0 = (const float*)d_in[1];   // (B, 1)    f32
    const float* ps = (const float*)d_in[2];   // (B, 4)    f32
    const int n = in_sizes[1];                 // B
    float* audio = (float*)d_out;              // (B, 64)
    float* pend  = audio + (size_t)n * NBLK;   // (B, 4)
    const int blocks = (n + TPB - 1) / TPB;
    fm_synth_kernel<<<blocks, TPB, 0, stream>>>(fm, f0, ps, audio, pend, n);
}